// DBLoss_11605001634022
// MI455X (gfx1250) — compile-verified
//
#include <hip/hip_runtime.h>
#include <hip/hip_bf16.h>

// ---------------------------------------------------------------------------
// DB loss (differentiable binarization): Ls + Lb + 10*Lt over 16x640x640 maps.
// Memory-bound streaming reduction (~105MB @ 23.3TB/s => ~4.5us floor).
// Exact hard-negative mining via device-side radix-select (flag-gated; with
// the reference input distribution all negatives are kept, so the select
// kernels early-exit). WMMA f32 16x16x4 used for the wave-level reduction.
// ---------------------------------------------------------------------------

typedef float v2f __attribute__((ext_vector_type(2)));
typedef float v8f __attribute__((ext_vector_type(8)));

// Workspace layout (as u32/f32 slots in d_ws)
enum {
    NPOS_S = 0,     // u32 atomic
    NPOS_B = 1,     // u32 atomic
    FPOS_SUM_S = 2, // f32 atomic
    FNEG_SUM_S = 3, // f32 atomic
    FPOS_SUM_B = 4, // f32 atomic
    FNEG_SUM_B = 5, // f32 atomic
    FLT_SUM = 6,    // f32 atomic
    FLAG_S = 7,     // u32: 1 => need top-k selection for Ls negatives
    FLAG_B = 8,
    ST_S = 9,       // [needk(u32), accsum(f32 bits), prefix(u32)]
    ST_B = 12,
    FTOP_S = 15,    // f32: selected negative sum
    FTOP_B = 16,
    NSEL_S = 17,    // u32: n_negative kept
    NSEL_B = 18,
    HIST_C = 32,            // u32[2048]
    HIST_S = 32 + 2048      // f32[2048]
};

__device__ __forceinline__ float bce_logits(float x, float t) {
    // max(x,0) - x*t + log1p(exp(-|x|)), valid for soft (even out-of-range) t
    return fmaxf(x, 0.0f) - x * t + log1pf(expf(-fabsf(x)));
}

// Full-wave (32-lane) f32 sum through the matrix pipe:
// A[m][k]: lanes0-15 vgpr0->k=0, lanes16-31 vgpr0->k=2 (vgpr1 zeroed).
// B = ones (layout invariant) => D[m][*] = p_m + p_{m+16}.
// Sum of 8 D vgprs per lane + xor-16 shuffle = total. EXEC must be all ones.
__device__ __forceinline__ float wave_sum_wmma(float p) {
    v2f a; a[0] = p;    a[1] = 0.0f;
    v2f b; b[0] = 1.0f; b[1] = 1.0f;
    v8f c = {0.f,0.f,0.f,0.f,0.f,0.f,0.f,0.f};
    c = __builtin_amdgcn_wmma_f32_16x16x4_f32(false, a, false, b, (short)0, c,
                                              false, false);
    float t = c[0] + c[1] + c[2] + c[3] + c[4] + c[5] + c[6] + c[7];
    t += __shfl_xor(t, 16, 32);
    return t;
}

__device__ __forceinline__ unsigned wave_sum_u32(unsigned v) {
    #pragma unroll
    for (int o = 16; o > 0; o >>= 1) v += __shfl_xor(v, o, 32);
    return v;
}

__device__ __forceinline__ void accum_elem(float p, float th, float tp, float tt,
                                           float& aps, float& ans, float& apb,
                                           float& anb, float& alt,
                                           unsigned& cps, unsigned& cpb) {
    float ls = bce_logits(p, tp);
    if (tp > 0.5f) { cps++; aps += ls; } else { ans += ls; }
    float bx = 50.0f * (p - th);
    float bt = 50.0f * (tp - tt);
    float lb = bce_logits(bx, bt);
    if (bt > 0.5f) { cpb++; apb += lb; } else { anb += lb; }
    alt += fabsf(th - tt);
}

__global__ void k_zero_state(unsigned* ws) {
    int i = threadIdx.x;
    if (i < 32) ws[i] = 0u;
}

__global__ void k_main(const float* __restrict__ p, const float* __restrict__ th,
                       const float* __restrict__ tp, const float* __restrict__ tt,
                       int n, unsigned* __restrict__ ws) {
    float aps = 0.f, ans = 0.f, apb = 0.f, anb = 0.f, alt = 0.f;
    unsigned cps = 0u, cpb = 0u;

    const int nv = n >> 2;
    const int tid = blockIdx.x * blockDim.x + threadIdx.x;
    const int stride = gridDim.x * blockDim.x;
    const float4* __restrict__ p4  = (const float4*)p;
    const float4* __restrict__ th4 = (const float4*)th;
    const float4* __restrict__ tp4 = (const float4*)tp;
    const float4* __restrict__ tt4 = (const float4*)tt;

    for (int i = tid; i < nv; i += stride) {
        float4 vp = p4[i], vh = th4[i], vq = tp4[i], vr = tt4[i];
        accum_elem(vp.x, vh.x, vq.x, vr.x, aps, ans, apb, anb, alt, cps, cpb);
        accum_elem(vp.y, vh.y, vq.y, vr.y, aps, ans, apb, anb, alt, cps, cpb);
        accum_elem(vp.z, vh.z, vq.z, vr.z, aps, ans, apb, anb, alt, cps, cpb);
        accum_elem(vp.w, vh.w, vq.w, vr.w, aps, ans, apb, anb, alt, cps, cpb);
    }
    if (tid == 0) { // scalar tail (n not multiple of 4)
        for (int i = nv << 2; i < n; ++i)
            accum_elem(p[i], th[i], tp[i], tt[i], aps, ans, apb, anb, alt, cps, cpb);
    }

    // Block reduction: wave-level via WMMA (floats) / shuffles (counts),
    // then one atomic per block per accumulator. EXEC is all-ones here.
    __shared__ float   sf[5][8];
    __shared__ unsigned sc[2][8];
    const int lane = threadIdx.x & 31;
    const int w    = threadIdx.x >> 5;

    float fv[5] = {aps, ans, apb, anb, alt};
    #pragma unroll
    for (int j = 0; j < 5; ++j) {
        float s = wave_sum_wmma(fv[j]);
        if (lane == 0) sf[j][w] = s;
    }
    unsigned cv[2] = {cps, cpb};
    #pragma unroll
    for (int j = 0; j < 2; ++j) {
        unsigned s = wave_sum_u32(cv[j]);
        if (lane == 0) sc[j][w] = s;
    }
    __syncthreads();
    if (threadIdx.x == 0) {
        float* wsf = (float*)ws;
        const int nw = (blockDim.x + 31) >> 5;
        float t0=0,t1=0,t2=0,t3=0,t4=0; unsigned u0=0,u1=0;
        for (int k = 0; k < nw; ++k) {
            t0+=sf[0][k]; t1+=sf[1][k]; t2+=sf[2][k]; t3+=sf[3][k]; t4+=sf[4][k];
            u0+=sc[0][k]; u1+=sc[1][k];
        }
        atomicAdd(&wsf[FPOS_SUM_S], t0);
        atomicAdd(&wsf[FNEG_SUM_S], t1);
        atomicAdd(&wsf[FPOS_SUM_B], t2);
        atomicAdd(&wsf[FNEG_SUM_B], t3);
        atomicAdd(&wsf[FLT_SUM],    t4);
        atomicAdd(&ws[NPOS_S], u0);
        atomicAdd(&ws[NPOS_B], u1);
    }
}

__global__ void k_prep(int n, unsigned* ws) {
    unsigned np_s = ws[NPOS_S];
    unsigned nn_s = (unsigned)n - np_s;
    unsigned k_s  = 3u * np_s; if (k_s > nn_s) k_s = nn_s;
    ws[FLAG_S] = (k_s < nn_s) ? 1u : 0u;
    ws[ST_S + 0] = k_s; ws[ST_S + 1] = 0u; ws[ST_S + 2] = 0u;
    ws[NSEL_S] = k_s;

    unsigned np_b = ws[NPOS_B];
    unsigned nn_b = (unsigned)n - np_b;
    unsigned k_b  = 3u * np_b; if (k_b > nn_b) k_b = nn_b;
    ws[FLAG_B] = (k_b < nn_b) ? 1u : 0u;
    ws[ST_B + 0] = k_b; ws[ST_B + 1] = 0u; ws[ST_B + 2] = 0u;
    ws[NSEL_B] = k_b;
}

__global__ void k_zero_hist(unsigned* ws) {
    int i = blockIdx.x * blockDim.x + threadIdx.x;
    if (i < 2048) { ws[HIST_C + i] = 0u; ((float*)ws)[HIST_S + i] = 0.0f; }
}

// Monotonic float->u32 key (handles negative losses from soft targets)
__device__ __forceinline__ unsigned f2key(float f) {
    unsigned u = __float_as_uint(f);
    return (u & 0x80000000u) ? ~u : (u | 0x80000000u);
}

__global__ void k_hist(const float* __restrict__ p, const float* __restrict__ th,
                       const float* __restrict__ tp, const float* __restrict__ tt,
                       int n, int which, int pass, unsigned* __restrict__ ws) {
    if (!ws[FLAG_S + which]) return; // selection not needed: all negatives kept
    const unsigned prefix = ws[(which ? ST_B : ST_S) + 2];
    unsigned* gc = ws + HIST_C;
    float*    gs = (float*)ws + HIST_S;

    __shared__ unsigned hc[2048];
    __shared__ float    hs[2048];
    for (int i = threadIdx.x; i < 2048; i += blockDim.x) { hc[i] = 0u; hs[i] = 0.f; }
    __syncthreads();

    const int tid = blockIdx.x * blockDim.x + threadIdx.x;
    const int stride = gridDim.x * blockDim.x;
    for (int i = tid; i < n; i += stride) {
        float x, t;
        if (which == 0) { x = p[i]; t = tp[i]; }
        else { x = 50.0f * (p[i] - th[i]); t = 50.0f * (tp[i] - tt[i]); }
        if (t > 0.5f) continue; // positive class: excluded from mining
        float L = bce_logits(x, t);
        unsigned key = f2key(L);
        unsigned bin; bool ok;
        if (pass == 0)      { ok = true;                    bin = key >> 21; }
        else if (pass == 1) { ok = ((key >> 21) == prefix); bin = (key >> 10) & 0x7FFu; }
        else                { ok = ((key >> 10) == prefix); bin = key & 0x3FFu; }
        if (ok) { atomicAdd(&hc[bin], 1u); atomicAdd(&hs[bin], L); }
    }
    __syncthreads();
    for (int i = threadIdx.x; i < 2048; i += blockDim.x) {
        if (hc[i]) { atomicAdd(&gc[i], hc[i]); atomicAdd(&gs[i], hs[i]); }
    }
}

__global__ void k_scan(int pass, int which, unsigned* ws) {
    if (!ws[FLAG_S + which]) return;
    unsigned* st = ws + (which ? ST_B : ST_S);
    unsigned needk  = st[0];
    float    accsum = __uint_as_float(st[1]);
    unsigned prefix = st[2];
    const unsigned* gc = ws + HIST_C;
    const float*    gs = (const float*)ws + HIST_S;

    const int nb = (pass == 2) ? 1024 : 2048;
    unsigned cum = 0; int cut = 0;
    for (int b = nb - 1; b >= 0; --b) {
        unsigned c = gc[b];
        if (cum + c >= needk) { cut = b; break; }
        cum += c;
        accsum += gs[b];
    }
    needk -= cum;
    if (pass < 2) {
        prefix = (pass == 0) ? (unsigned)cut : ((prefix << 11) | (unsigned)cut);
        st[0] = needk; st[1] = __float_as_uint(accsum); st[2] = prefix;
    } else {
        unsigned key = (prefix << 10) | (unsigned)cut;
        unsigned u   = (key & 0x80000000u) ? (key & 0x7FFFFFFFu) : ~key;
        float v = __uint_as_float(u); // exact k-th largest negative loss
        ((float*)ws)[FTOP_S + which] = accsum + (float)needk * v;
    }
}

__global__ void k_final(int n, const unsigned* ws, float* out) {
    const float* wsf = (const float*)ws;
    float neg_s = ws[FLAG_S] ? wsf[FTOP_S] : wsf[FNEG_SUM_S];
    float neg_b = ws[FLAG_B] ? wsf[FTOP_B] : wsf[FNEG_SUM_B];
    float Ls = (wsf[FPOS_SUM_S] + neg_s) / (float)(ws[NPOS_S] + ws[NSEL_S]);
    float Lb = (wsf[FPOS_SUM_B] + neg_b) / (float)(ws[NPOS_B] + ws[NSEL_B]);
    float Lt = wsf[FLT_SUM] / (float)n;
    out[0] = Ls + Lb + 10.0f * Lt;
}

extern "C" void kernel_launch(void* const* d_in, const int* in_sizes, int n_in,
                              void* d_out, int out_size, void* d_ws, size_t ws_size,
                              hipStream_t stream) {
    const float* p  = (const float*)d_in[0]; // proba_map
    const float* th = (const float*)d_in[1]; // thresh_map
    const float* tp = (const float*)d_in[2]; // target_proba_map
    const float* tt = (const float*)d_in[3]; // target_thresh_map
    const int n = in_sizes[0];
    unsigned* ws = (unsigned*)d_ws;
    float* out = (float*)d_out;

    k_zero_state<<<1, 64, 0, stream>>>(ws);

    int nv = n >> 2;
    int blocks = (nv + 255) / 256;
    if (blocks > 2048) blocks = 2048;
    if (blocks < 1) blocks = 1;
    k_main<<<blocks, 256, 0, stream>>>(p, th, tp, tt, n, ws);

    k_prep<<<1, 1, 0, stream>>>(n, ws);

    // Exact top-k radix select (flag-gated; no-ops when all negatives kept)
    for (int pass = 0; pass < 3; ++pass) {
        for (int which = 0; which < 2; ++which) {
            k_zero_hist<<<8, 256, 0, stream>>>(ws);
            k_hist<<<1024, 256, 0, stream>>>(p, th, tp, tt, n, which, pass, ws);
            k_scan<<<1, 1, 0, stream>>>(pass, which, ws);
        }
    }

    k_final<<<1, 1, 0, stream>>>(n, ws, out);
}